// RecLayer_15779709845863
// MI455X (gfx1250) — compile-verified
//
#include <hip/hip_runtime.h>
#include <stdint.h>

#define DFEAT 64

// ---------------------------------------------------------------------------
// PROBE (first in .text so the disasm snippet shows its whole body; never
// launched): verifies that unsafeAtomicAdd lowers to the hardware
// global_atomic_add_f32 (not a CAS loop) and that __builtin_prefetch emits
// global_prefetch_b8 on gfx1250.
// ---------------------------------------------------------------------------
__global__ void gcn_probe(float* __restrict__ p, const float* __restrict__ v) {
    int i = threadIdx.x;
    __builtin_prefetch(p + i + 2048, 0, 3);
    unsafeAtomicAdd(p + i, v[i]);
}

// ---------------------------------------------------------------------------
// MAIN kernel: fused gather -> scale -> atomic scatter.  One wave32 per edge,
// grid-stride over edges.  Lane l handles feature elements l and l+32, so
// every load and every atomic instruction is a contiguous 128B burst across
// the wave.  The in-degree normalization is folded into the per-edge scale
// (rst * c == sum(msg) * c).  The next edge's source row is prefetched
// (global_prefetch_b8) to hide the data-dependent L2 gather latency, and the
// f32 adds use the hardware global_atomic_add_f32 path (unsafeAtomicAdd,
// non-returning -> STOREcnt), not a CAS loop.
// ---------------------------------------------------------------------------
__global__ void gcn_edge_scatter(const float* __restrict__ h_src,
                                 const float* __restrict__ mask,
                                 const int*  __restrict__ src_idx,
                                 const int*  __restrict__ dst_idx,
                                 const float* __restrict__ out_norm,  // [n_src]
                                 const float* __restrict__ in_norm,   // [n_dst]
                                 float* __restrict__ out,             // [n_dst*64]
                                 long E) {
    const int  lane = threadIdx.x & 31;
    const long wid  = (long)blockIdx.x * (blockDim.x >> 5) + (threadIdx.x >> 5);
    const long nw   = (long)gridDim.x * (blockDim.x >> 5);

    for (long e = wid; e < E; e += nw) {
        // Prefetch the next iteration's source row: 32 lanes x 1 dword touch
        // only the first 128B of the 256B row.
        long en = e + nw;
        if (en < E) {
            int sn = src_idx[en];
            __builtin_prefetch(h_src + (size_t)sn * DFEAT + lane, 0, 3);
        }

        // Wave-uniform edge scalars (one cacheline request per wave).
        int   s = src_idx[e];
        int   t = dst_idx[e];
        float scale = mask[e] * out_norm[s] * in_norm[t];

        const float* row  = h_src + (size_t)s * DFEAT;
        float a = row[lane];
        float b = row[lane + 32];

        float* orow = out + (size_t)t * DFEAT;
        unsafeAtomicAdd(&orow[lane],      scale * a);
        unsafeAtomicAdd(&orow[lane + 32], scale * b);
    }
}

// ---------------------------------------------------------------------------
// Zero a buffer with wide b128 stores (bulk is the 25.6MB output clear).
// ---------------------------------------------------------------------------
__global__ void gcn_zero_b128(uint4* __restrict__ p, long n128) {
    long i      = (long)blockIdx.x * blockDim.x + threadIdx.x;
    long stride = (long)gridDim.x * blockDim.x;
    uint4 z; z.x = 0u; z.y = 0u; z.z = 0u; z.w = 0u;
    for (; i < n128; i += stride) p[i] = z;
}

__global__ void gcn_zero_u32(uint32_t* __restrict__ p, long n) {
    long i      = (long)blockIdx.x * blockDim.x + threadIdx.x;
    long stride = (long)gridDim.x * blockDim.x;
    for (; i < n; i += stride) p[i] = 0u;
}

// ---------------------------------------------------------------------------
// Integer degree histograms: outdeg at deg[0..n_src), indeg at
// deg[n_src..n_src+n_dst).  global_atomic_add_u32 traffic lands in the 192MB
// L2 (800KB working set).
// ---------------------------------------------------------------------------
__global__ void gcn_degree(const int* __restrict__ src_idx,
                           const int* __restrict__ dst_idx,
                           uint32_t* __restrict__ outdeg,
                           uint32_t* __restrict__ indeg,
                           long E) {
    long i      = (long)blockIdx.x * blockDim.x + threadIdx.x;
    long stride = (long)gridDim.x * blockDim.x;
    for (; i < E; i += stride) {
        atomicAdd(&outdeg[src_idx[i]], 1u);
        atomicAdd(&indeg[dst_idx[i]], 1u);
    }
}

// ---------------------------------------------------------------------------
// deg -> 1/sqrt(max(deg,1)) in place.  All accesses through the uint32_t*
// type (float stored via bit cast) to stay aliasing-clean.
// ---------------------------------------------------------------------------
__global__ void gcn_norm(uint32_t* __restrict__ p, long n) {
    long i = (long)blockIdx.x * blockDim.x + threadIdx.x;
    if (i < n) {
        uint32_t d  = p[i];
        float    df = (float)(d > 0u ? d : 1u);
        p[i] = __float_as_uint(1.0f / sqrtf(df));
    }
}

// ---------------------------------------------------------------------------
// Host-side launcher.  Inputs (setup_inputs order):
//   d_in[0] h_src   float32 [n_src*64]
//   d_in[1] mask    float32 [E]
//   d_in[2] src_idx int32   [E]
//   d_in[3] dst_idx int32   [E]
//   d_in[4] n_dst   int32   [1]   (recovered from out_size instead)
// d_out: float32 [n_dst*64]
// d_ws layout: uint32 deg/norm array of (n_src + n_dst) entries.
// ---------------------------------------------------------------------------
extern "C" void kernel_launch(void* const* d_in, const int* in_sizes, int n_in,
                              void* d_out, int out_size, void* d_ws, size_t ws_size,
                              hipStream_t stream) {
    const float* h_src   = (const float*)d_in[0];
    const float* mask    = (const float*)d_in[1];
    const int*   src_idx = (const int*)d_in[2];
    const int*   dst_idx = (const int*)d_in[3];

    const long n_src = (long)in_sizes[0] / DFEAT;
    const long E     = (long)in_sizes[1];
    const long n_dst = (long)out_size / DFEAT;

    float*    out = (float*)d_out;
    uint32_t* deg = (uint32_t*)d_ws;          // [n_src + n_dst] u32 -> f32 norms
    const long n_nodes = n_src + n_dst;

    // 1) Zero degree scratch and output (harness poisons them with 0xAA).
    {
        int blk = 256;
        int g1  = (int)((n_nodes + blk - 1) / blk); if (g1 > 2048) g1 = 2048;
        gcn_zero_u32<<<g1, blk, 0, stream>>>(deg, n_nodes);

        // Output is n_dst*64 floats: always a multiple of 4 dwords -> b128.
        long n128 = (n_dst * DFEAT) / 4;
        int g2 = (int)((n128 + blk - 1) / blk); if (g2 > 4096) g2 = 4096;
        gcn_zero_b128<<<g2, blk, 0, stream>>>((uint4*)out, n128);
    }

    // 2) Degree histograms.
    {
        int blk = 256;
        int g = (int)((E + blk - 1) / blk); if (g > 4096) g = 4096;
        gcn_degree<<<g, blk, 0, stream>>>(src_idx, dst_idx, deg, deg + n_src, E);
    }

    // 3) deg -> 1/sqrt(max(deg,1)).
    {
        int blk = 256;
        int g = (int)((n_nodes + blk - 1) / blk);
        gcn_norm<<<g, blk, 0, stream>>>(deg, n_nodes);
    }

    // 4) Fused edge gather/scale/scatter: one wave32 per edge, grid-stride.
    {
        int blk = 256;                       // 8 waves per block
        long waves_needed = (E + 18) / 19;   // ~19 edges per wave
        int g = (int)((waves_needed + 7) / 8);
        if (g > 8192) g = 8192;
        if (g < 1)    g = 1;
        gcn_edge_scatter<<<g, blk, 0, stream>>>(
            h_src, mask, src_idx, dst_idx,
            (const float*)deg,                 // out_norm
            (const float*)(deg + n_src),       // in_norm
            out, E);
    }
}